// DeformRoIPool_13469017440354
// MI455X (gfx1250) — compile-verified
//
#include <hip/hip_runtime.h>
#include <stdint.h>

// Problem constants from the reference
#define PH_   7
#define PW_   7
#define CH_   256
#define HH_   128
#define WW_   128
#define HW_   (HH_ * WW_)
#define NBIN  (PH_ * PW_)            // 49
#define NSAMP (NBIN * 4)             // 196 (49 bins x 2x2 sub-samples)
#define CPB   128                    // channels per block (grid.y = 2)
#define CSTRIDE 32                   // channel quad: c, c+32, c+64, c+96
#define QUADN (CSTRIDE * NBIN)       // 1568 thread-work items per block
#define BLK   224                    // 7 wave32s; 1568 / 224 == 7 exact iterations

// 8-byte load from a 4-byte-aligned address (lowers to global_load_b64)
__device__ __forceinline__ float2 ld2(const float* p) {
    float2 v;
    __builtin_memcpy(&v, p, 8);
    return v;
}

__global__ __launch_bounds__(BLK) void deform_roi_pool_kernel(
    const float* __restrict__ feat,   // [B, C, H, W]
    const float* __restrict__ rois,   // [R, 5]
    const float* __restrict__ offs,   // [R, 2, 7, 7]
    float* __restrict__ out)          // [R, C, 7, 7]
{
    __shared__ float  s_roi[8];
    __shared__ float  s_off[98];
    __shared__ __align__(16) int s_o[NSAMP];   // row0 offset per sample (py*W+px)
    __shared__ float4 s_w[NSAMP];              // 4 corner weights (premult valid*0.25)

    const int tid   = threadIdx.x;
    const int r     = blockIdx.x;
    const int cbase = blockIdx.y * CPB;

    // ---- Phase 0: CDNA5 async global->LDS staging of roi row + offset slab ----
    if (tid < 103) {
        unsigned lds;
        unsigned long long ga;
        if (tid < 98) {
            lds = (unsigned)(uintptr_t)&s_off[tid];
            ga  = (unsigned long long)(uintptr_t)(offs + (size_t)r * 98 + tid);
        } else {
            lds = (unsigned)(uintptr_t)&s_roi[tid - 98];
            ga  = (unsigned long long)(uintptr_t)(rois + (size_t)r * 5 + (tid - 98));
        }
        asm volatile("global_load_async_to_lds_b32 %0, %1, off"
                     :: "v"(lds), "v"(ga) : "memory");
    }
    asm volatile("s_wait_asynccnt 0x0" ::: "memory");
    __syncthreads();

    // ---- Phase 1: per-sample geometry (196 samples), mmcv bilinear semantics ----
    if (tid < NSAMP) {
        const int bin = tid >> 2;
        const int s   = tid & 3;
        const int ph  = bin / PW_;
        const int pw  = bin - ph * PW_;
        const int iy  = s >> 1;
        const int ix  = s & 1;

        const float x1   = s_roi[1] * 0.25f - 0.5f;
        const float y1   = s_roi[2] * 0.25f - 0.5f;
        const float roiw = (s_roi[3] - s_roi[1]) * 0.25f;
        const float roih = (s_roi[4] - s_roi[2]) * 0.25f;
        const float binw = roiw / 7.0f;
        const float binh = roih / 7.0f;

        float x = x1 + 0.1f * roiw * s_off[bin]
                     + (float)pw * binw
                     + ((float)ix + 0.5f) * 0.5f * binw;
        float y = y1 + 0.1f * roih * s_off[49 + bin]
                     + (float)ph * binh
                     + ((float)iy + 0.5f) * 0.5f * binh;

        const bool valid = (y >= -1.0f) && (y <= (float)HH_) &&
                           (x >= -1.0f) && (x <= (float)WW_);
        const float yc = fmaxf(y, 0.0f);
        const float xc = fmaxf(x, 0.0f);
        const int yl = (int)yc;            // yc >= 0 -> trunc == floor
        const int xl = (int)xc;

        // Bilinear over rows (py,py+1) x cols (px,px+1); clamping folded into
        // the weights (clamped => all weight on the far element).
        int py, px;
        float wya, wyb, wxa, wxb;
        if (yl >= HH_ - 1) { py = HH_ - 2; wya = 0.0f; wyb = 1.0f; }
        else { py = yl; const float ly = yc - (float)yl; wyb = ly; wya = 1.0f - ly; }
        if (xl >= WW_ - 1) { px = WW_ - 2; wxa = 0.0f; wxb = 1.0f; }
        else { px = xl; const float lx = xc - (float)xl; wxb = lx; wxa = 1.0f - lx; }

        const float vs = valid ? 0.25f : 0.0f;   // fold 1/(S*S) average + validity
        s_o[tid] = py * WW_ + px;
        s_w[tid] = make_float4(vs * wya * wxa, vs * wya * wxb,
                               vs * wyb * wxa, vs * wyb * wxb);
    }
    __syncthreads();

    // ---- Phase 2: 4 channels per thread (c, c+32, c+64, c+96); one vaddr per
    //      sample, 8x b64 loads with immediate offsets; coalesced NT stores ----
    const int b = (int)s_roi[0];
    const float* fb = feat + (size_t)b * ((size_t)CH_ * HW_);
    const size_t outbase = (size_t)r * ((size_t)CH_ * NBIN) + (size_t)cbase * NBIN;

    #pragma unroll 1
    for (int it = 0; it < QUADN / BLK; ++it) {          // exactly 7 iterations
        const int flat  = it * BLK + tid;               // [0, 1568)
        const int cquad = flat / NBIN;                  // [0, 32)
        const int bin   = flat - cquad * NBIN;          // [0, 49)

        const float* base = fb + (size_t)(cbase + cquad) * HW_;
        const int4 o = ((const int4*)s_o)[bin];         // 4 sample offsets (ds_load_b128)
        const int p0 = bin << 2;
        float acc0 = 0.0f, acc1 = 0.0f, acc2 = 0.0f, acc3 = 0.0f;
        #pragma unroll
        for (int s = 0; s < 4; ++s) {
            const int    os = (s == 0) ? o.x : (s == 1) ? o.y : (s == 2) ? o.z : o.w;
            const float4 w  = s_w[p0 + s];
            const float* bp = base + os;                // single 64-bit address calc
            // 4 channel slabs at constant byte offsets 0 / 2MB / 4MB / 6MB,
            // each needing rows py (imm +0) and py+1 (imm +512B)
            const float2 t0 = ld2(bp);
            const float2 u0 = ld2(bp + WW_);
            const float2 t1 = ld2(bp + 1 * CSTRIDE * HW_);
            const float2 u1 = ld2(bp + 1 * CSTRIDE * HW_ + WW_);
            const float2 t2 = ld2(bp + 2 * CSTRIDE * HW_);
            const float2 u2 = ld2(bp + 2 * CSTRIDE * HW_ + WW_);
            const float2 t3 = ld2(bp + 3 * CSTRIDE * HW_);
            const float2 u3 = ld2(bp + 3 * CSTRIDE * HW_ + WW_);
            acc0 = fmaf(w.x, t0.x, acc0);
            acc0 = fmaf(w.y, t0.y, acc0);
            acc0 = fmaf(w.z, u0.x, acc0);
            acc0 = fmaf(w.w, u0.y, acc0);
            acc1 = fmaf(w.x, t1.x, acc1);
            acc1 = fmaf(w.y, t1.y, acc1);
            acc1 = fmaf(w.z, u1.x, acc1);
            acc1 = fmaf(w.w, u1.y, acc1);
            acc2 = fmaf(w.x, t2.x, acc2);
            acc2 = fmaf(w.y, t2.y, acc2);
            acc2 = fmaf(w.z, u2.x, acc2);
            acc2 = fmaf(w.w, u2.y, acc2);
            acc3 = fmaf(w.x, t3.x, acc3);
            acc3 = fmaf(w.y, t3.y, acc3);
            acc3 = fmaf(w.z, u3.x, acc3);
            acc3 = fmaf(w.w, u3.y, acc3);
        }
        __builtin_nontemporal_store(acc0, out + outbase + (size_t)flat);
        __builtin_nontemporal_store(acc1, out + outbase + (size_t)(flat + 1 * QUADN));
        __builtin_nontemporal_store(acc2, out + outbase + (size_t)(flat + 2 * QUADN));
        __builtin_nontemporal_store(acc3, out + outbase + (size_t)(flat + 3 * QUADN));
    }
}

extern "C" void kernel_launch(void* const* d_in, const int* in_sizes, int n_in,
                              void* d_out, int out_size, void* d_ws, size_t ws_size,
                              hipStream_t stream) {
    const float* feat = (const float*)d_in[0];   // [2, 256, 128, 128]
    const float* rois = (const float*)d_in[1];   // [R, 5]
    const float* offs = (const float*)d_in[2];   // [R, 2, 7, 7]
    float* out = (float*)d_out;                  // [R, 256, 7, 7]

    const int R = in_sizes[1] / 5;               // 512
    dim3 grid((unsigned)R, CH_ / CPB);           // (512, 2)
    deform_roi_pool_kernel<<<grid, BLK, 0, stream>>>(feat, rois, offs, out);
}